// Fusionmodel_9912784520058
// MI455X (gfx1250) — compile-verified
//
#include <hip/hip_runtime.h>

// ---------------- problem constants ----------------
#define Bsz 8
#define Cch 256
#define C2  128
#define Hd  96
#define Wd  96
#define HW  (Hd*Wd)             // 9216  (multiple of 16; W=96 multiple of 16)
#define Mrows (Bsz*HW)          // 73728
#define K3  (3*Cch)             // 768
#define NTOT ((long)Bsz*Cch*HW) // 18874368
#define PU  144                 // 12*12 pixels per patch
#define NP  64                  // 8*8 patches
#define KC  32                  // K chunk staged in LDS

typedef float v2f __attribute__((ext_vector_type(2)));
typedef float v8f __attribute__((ext_vector_type(8)));

__device__ __forceinline__ v8f wmma_f32x4(v2f a, v2f b, v8f c) {
  // D = A(16x4 f32) * B(4x16 f32) + C(16x16 f32)
  return __builtin_amdgcn_wmma_f32_16x16x4_f32(false, a, false, b, (short)0, c, false, false);
}

// CDNA5 async copy: global -> LDS, tracked by ASYNCcnt
__device__ __forceinline__ void async_ld_f32(float* lds, const float* gp) {
  unsigned int lo = (unsigned int)(unsigned long long)lds;  // LDS offset = low 32 bits of flat addr
  asm volatile("global_load_async_to_lds_b32 %0, %1, off" :: "v"(lo), "v"(gp) : "memory");
}
__device__ __forceinline__ void wait_async() {
  asm volatile("s_wait_asynccnt 0x0" ::: "memory");
}

// ---------------- 2x2 max pool ----------------
__global__ void pool2_kernel(const float* __restrict__ x, float* __restrict__ y,
                             int hin, int win, long nbc) {
  int ho = hin >> 1, wo = win >> 1;
  long total = nbc * ho * wo;
  long i = blockIdx.x * (long)blockDim.x + threadIdx.x;
  if (i >= total) return;
  int wv = (int)(i % wo); long t = i / wo;
  int hv = (int)(t % ho); long bc = t / ho;
  const float* p = x + (bc * hin + hv * 2) * win + wv * 2;
  y[i] = fmaxf(fmaxf(p[0], p[1]), fmaxf(p[win], p[win + 1]));
}

// ---------------- msc: relu(conv1x1([x|up(mp2)|up(mp4)], w) + b) ----------------
// 128 threads (4 waves): 16 rows x 64 cols per block; A chunk staged in LDS once.
__global__ void gemm_msc_kernel(const float* __restrict__ x, const float* __restrict__ m2,
                                const float* __restrict__ m4, const float* __restrict__ wmat,
                                const float* __restrict__ bias, float* __restrict__ feat) {
  __shared__ float As[KC * 16];   // [kk][m]
  int rowBase = blockIdx.x * 16;
  int b  = rowBase / HW;          // block-uniform (tile never crosses batch: HW%16==0)
  int r0 = rowBase % HW;          // block-uniform
  int h  = r0 / Wd;               // block-uniform (tile never crosses image row: Wd%16==0)
  int w0 = r0 % Wd;               // block-uniform
  int tid = threadIdx.x;
  int wave = tid >> 5, lane = tid & 31, lhalf = lane >> 4, l15 = lane & 15;
  int colBase = blockIdx.y * 64 + wave * 16;
  const float* wp  = wmat + (long)(colBase + l15) * K3;
  const float* xp  = x  + ((long)b * Cch) * HW + r0;                          // + c*HW + m
  const float* m2p = m2 + ((long)b * Cch) * 2304 + (h >> 1) * 48;             // + c*2304 + (w0+m)>>1
  const float* m4p = m4 + ((long)b * Cch) * 576  + (h >> 2) * 24;             // + c*576  + (w0+m)>>2

  int te_k = tid >> 4;            // base kk for this thread's staging slots
  int te_m = tid & 15;

  v8f acc = {};
  for (int k0 = 0; k0 < K3; k0 += KC) {
    // ---- stage A chunk: uniform source per chunk (256/512 boundaries are KC-aligned),
    //      fully unrolled (4 slots/thread) -> no exec-mask games around the async ops.
    if (k0 < Cch) {
#pragma unroll
      for (int it = 0; it < 4; ++it) {
        int kk = te_k + it * 8;
        async_ld_f32(&As[kk * 16 + te_m], xp + (long)(k0 + kk) * HW + te_m);
      }
    } else if (k0 < 2 * Cch) {
#pragma unroll
      for (int it = 0; it < 4; ++it) {
        int kk = te_k + it * 8;
        async_ld_f32(&As[kk * 16 + te_m], m2p + (long)(k0 - Cch + kk) * 2304 + ((w0 + te_m) >> 1));
      }
    } else {
#pragma unroll
      for (int it = 0; it < 4; ++it) {
        int kk = te_k + it * 8;
        async_ld_f32(&As[kk * 16 + te_m], m4p + (long)(k0 - 2 * Cch + kk) * 576 + ((w0 + te_m) >> 2));
      }
    }
    wait_async();
    __syncthreads();
    // ---- WMMA over the chunk ----
#pragma unroll
    for (int kk = 0; kk < KC; kk += 4) {
      int kq = kk + lhalf * 2;
      v2f av = { As[kq * 16 + l15], As[(kq + 1) * 16 + l15] };
      v2f bv = { wp[k0 + kq], wp[k0 + kq + 1] };
      acc = wmma_f32x4(av, bv, acc);
    }
    __syncthreads();
  }
  int o = colBase + l15;
  float bia = bias[o];
#pragma unroll
  for (int rr = 0; rr < 8; ++rr) {
    int m = rr + lhalf * 8;
    float v = acc[rr] + bia;
    v = v > 0.f ? v : 0.f;
    feat[((long)b * Cch + o) * HW + r0 + m] = v;
  }
}

// ---------------- generic conv1x1 GEMM with epilogue modes ----------------
// mode 0: out = dot + bias
// mode 1: out = relu((dot+bias)/sqrt(1+eps)*g + beta)
// mode 2: out = 0.5*(orig + dot + bias)
__global__ void gemm_conv1x1_kernel(const float* __restrict__ A, int Cin,
                                    const float* __restrict__ wmat, const float* __restrict__ bias,
                                    const float* __restrict__ g, const float* __restrict__ beta,
                                    const float* __restrict__ orig,
                                    float* __restrict__ out, int Cout, int mode) {
  __shared__ float As[KC * 16];   // [kk][m]
  int rowBase = blockIdx.x * 16;
  int b  = rowBase / HW;
  int r0 = rowBase % HW;
  int tid = threadIdx.x;
  int wave = tid >> 5, lane = tid & 31, lhalf = lane >> 4, l15 = lane & 15;
  int colBase = blockIdx.y * 64 + wave * 16;
  const float* wp = wmat + (long)(colBase + l15) * Cin;
  const float* ap = A + ((long)b * Cin) * HW + r0;   // + c*HW + m

  int te_k = tid >> 4;
  int te_m = tid & 15;

  v8f acc = {};
  for (int k0 = 0; k0 < Cin; k0 += KC) {
#pragma unroll
    for (int it = 0; it < 4; ++it) {
      int kk = te_k + it * 8;
      async_ld_f32(&As[kk * 16 + te_m], ap + (long)(k0 + kk) * HW + te_m);
    }
    wait_async();
    __syncthreads();
#pragma unroll
    for (int kk = 0; kk < KC; kk += 4) {
      int kq = kk + lhalf * 2;
      v2f av = { As[kq * 16 + l15], As[(kq + 1) * 16 + l15] };
      v2f bv = { wp[k0 + kq], wp[k0 + kq + 1] };
      acc = wmma_f32x4(av, bv, acc);
    }
    __syncthreads();
  }
  const float inv_sqrt = 0.99999500003749971f;  // 1/sqrt(1+1e-5)
  int o = colBase + l15;
  float bia = bias[o];
  float gg = (mode == 1) ? g[o] * inv_sqrt : 0.f;
  float bb = (mode == 1) ? beta[o] : 0.f;
#pragma unroll
  for (int rr = 0; rr < 8; ++rr) {
    int m = rr + lhalf * 8;
    long oi = ((long)b * Cout + o) * HW + r0 + m;
    float v = acc[rr] + bia;
    if (mode == 1)      { v = fmaxf(v * gg + bb, 0.f); }
    else if (mode == 2) { v = 0.5f * (orig[oi] + v); }
    out[oi] = v;
  }
}

// ---------------- patch attention: softmax(q^T k / sqrt(C2)) @ v^T ----------------
// one 256-thread block (8 waves) per (b, pixel-in-patch) pair. 1152 blocks.
__global__ void attn_kernel(const float* __restrict__ q, const float* __restrict__ k,
                            const float* __restrict__ v, float* __restrict__ ctx) {
  __shared__ float qs[C2 * NP];        // 32 KB, [c][n]
  __shared__ float S[NP * (NP + 1)];   // padded score rows

  int bp = blockIdx.x;
  int b = bp / PU; int ru = bp % PU; int rh = ru / 12; int rw = ru % 12;
  int tid = threadIdx.x;

  // stage q tile into LDS (literal trip count: 8192/256 = 32)
#pragma unroll 4
  for (int it = 0; it < (C2 * NP) / 256; ++it) {
    int e = tid + it * 256;
    int c = e >> 6, n = e & 63;
    int ih = n >> 3, iw = n & 7;
    qs[e] = q[(((long)b * C2 + c) * Hd + ih * 12 + rh) * Wd + iw * 12 + rw];
  }
  __syncthreads();

  int wave = tid >> 5, lane = tid & 31, lhalf = lane >> 4, l15 = lane & 15;

  // S[n][m] = sum_c q[c][n]*k[c][m] * C2^-0.5 ; 4x4 grid of 16x16 tiles
#pragma unroll
  for (int tt = 0; tt < 2; ++tt) {
    int t = wave + tt * 8;
    int tn = (t >> 2) * 16, tm = (t & 3) * 16;
    int mcol = tm + l15;
    int mih = mcol >> 3, miw = mcol & 7;
    const float* kp = k + (((long)b * C2) * Hd + mih * 12 + rh) * Wd + miw * 12 + rw;
    v8f acc = {};
    for (int kb = 0; kb < C2; kb += 4) {
      int kk = kb + lhalf * 2;
      v2f av = { qs[kk * NP + tn + l15], qs[(kk + 1) * NP + tn + l15] };
      v2f bv = { kp[(long)kk * HW], kp[(long)(kk + 1) * HW] };
      acc = wmma_f32x4(av, bv, acc);
    }
#pragma unroll
    for (int rr = 0; rr < 8; ++rr) {
      int n = tn + rr + lhalf * 8;
      S[n * (NP + 1) + tm + l15] = acc[rr] * 0.08838834764831845f; // 128^-0.5
    }
  }
  __syncthreads();

  // row softmax (64 rows, one thread each)
  if (tid < NP) {
    float* row = S + tid * (NP + 1);
    float mx = row[0];
    for (int i = 1; i < NP; ++i) mx = fmaxf(mx, row[i]);
    float sum = 0.f;
    for (int i = 0; i < NP; ++i) { float e = expf(row[i] - mx); row[i] = e; sum += e; }
    float inv = 1.f / sum;
    for (int i = 0; i < NP; ++i) row[i] *= inv;
  }
  __syncthreads();

  // ctx[n][c] = sum_m S[n][m] * v[c][m] ; 4x8 grid of 16x16 tiles, 4 per wave
#pragma unroll
  for (int tt = 0; tt < 4; ++tt) {
    int t = wave * 4 + tt;
    int tn = (t >> 3) * 16, tc = (t & 7) * 16;
    int ccol = tc + l15;
    const float* vp = v + ((long)b * C2 + ccol) * HW + rh * Wd + rw;
    v8f acc = {};
    for (int kb = 0; kb < NP; kb += 4) {
      int kk = kb + lhalf * 2;
      v2f av = { S[(tn + l15) * (NP + 1) + kk], S[(tn + l15) * (NP + 1) + kk + 1] };
      int ih0 = kk >> 3, iw0 = kk & 7;
      int ih1 = (kk + 1) >> 3, iw1 = (kk + 1) & 7;
      v2f bv = { vp[(ih0 * 12) * Wd + iw0 * 12], vp[(ih1 * 12) * Wd + iw1 * 12] };
      acc = wmma_f32x4(av, bv, acc);
    }
#pragma unroll
    for (int rr = 0; rr < 8; ++rr) {
      int n = tn + rr + lhalf * 8;
      int c = tc + l15;
      int ih = n >> 3, iw = n & 7;
      ctx[(((long)b * C2 + c) * Hd + ih * 12 + rh) * Wd + iw * 12 + rw] = acc[rr];
    }
  }
}

// ---------------- fusion = (new_RGB + new_D)/2 ----------------
__global__ void fuse_kernel(const float* __restrict__ a, const float* __restrict__ b,
                            float* __restrict__ o, long n) {
  long i = blockIdx.x * (long)blockDim.x + threadIdx.x;
  if (i < n) o[i] = 0.5f * (a[i] + b[i]);
}

// ---------------- launch ----------------
extern "C" void kernel_launch(void* const* d_in, const int* in_sizes, int n_in,
                              void* d_out, int out_size, void* d_ws, size_t ws_size,
                              hipStream_t stream) {
  (void)in_sizes; (void)n_in; (void)out_size; (void)ws_size;
  const float* RGB = (const float*)d_in[0];
  const float* Din = (const float*)d_in[1];
  float* out = (float*)d_out;

  float* ws   = (float*)d_ws;
  float* m2   = ws;                        // 4,718,592
  float* m4   = m2 + 4718592;              // 1,179,648
  float* feat = m4 + 1179648;              // 18,874,368
  float* qb   = feat + 18874368;           // 9,437,184
  float* kb   = qb + 9437184;
  float* vb   = kb + 9437184;
  float* ctx  = vb + 9437184;              // total ~238.5 MB

  for (int br = 0; br < 2; ++br) {
    const float* x = br ? Din : RGB;
    int pi = 2 + br * 14;
    const float* msc_w  = (const float*)d_in[pi + 0];
    const float* msc_b  = (const float*)d_in[pi + 1];
    const float* q_w    = (const float*)d_in[pi + 2];
    const float* q_b    = (const float*)d_in[pi + 3];
    const float* k_w    = (const float*)d_in[pi + 4];
    const float* k_b    = (const float*)d_in[pi + 5];
    const float* v_w    = (const float*)d_in[pi + 6];
    const float* v_b    = (const float*)d_in[pi + 7];
    const float* q_g    = (const float*)d_in[pi + 8];
    const float* q_beta = (const float*)d_in[pi + 9];
    const float* k_g    = (const float*)d_in[pi + 10];
    const float* k_beta = (const float*)d_in[pi + 11];
    const float* W_w    = (const float*)d_in[pi + 12];
    const float* W_b    = (const float*)d_in[pi + 13];

    long n2 = (long)Bsz * Cch * 48 * 48;
    long n4 = (long)Bsz * Cch * 24 * 24;
    pool2_kernel<<<dim3((unsigned)((n2 + 255) / 256)), 256, 0, stream>>>(x, m2, Hd, Wd, (long)Bsz * Cch);
    pool2_kernel<<<dim3((unsigned)((n4 + 255) / 256)), 256, 0, stream>>>(m2, m4, 48, 48, (long)Bsz * Cch);

    gemm_msc_kernel<<<dim3(Mrows / 16, Cch / 64), 128, 0, stream>>>(x, m2, m4, msc_w, msc_b, feat);

    dim3 gq(Mrows / 16, C2 / 64);    // 4608 x 2
    gemm_conv1x1_kernel<<<gq, 128, 0, stream>>>(feat, Cch, q_w, q_b, q_g, q_beta, nullptr, qb, C2, 1);
    gemm_conv1x1_kernel<<<gq, 128, 0, stream>>>(feat, Cch, k_w, k_b, k_g, k_beta, nullptr, kb, C2, 1);
    gemm_conv1x1_kernel<<<gq, 128, 0, stream>>>(feat, Cch, v_w, v_b, nullptr, nullptr, nullptr, vb, C2, 0);

    attn_kernel<<<dim3(Bsz * PU), 256, 0, stream>>>(qb, kb, vb, ctx);

    gemm_conv1x1_kernel<<<dim3(Mrows / 16, Cch / 64), 128, 0, stream>>>(ctx, C2, W_w, W_b, nullptr, nullptr, x,
                                                                        out + (long)br * NTOT, Cch, 2);
  }
  fuse_kernel<<<dim3((unsigned)((NTOT + 255) / 256)), 256, 0, stream>>>(out, out + NTOT,
                                                                        out + 2 * NTOT, NTOT);
}